// LaplacianLoss_7997229105177
// MI455X (gfx1250) — compile-verified
//
#include <hip/hip_runtime.h>

typedef __attribute__((ext_vector_type(2))) float v2f;
typedef __attribute__((ext_vector_type(8))) float v8f;

#define BB 4
#define NN 50000
#define DD 64
#define KK 16

// ---------------------------------------------------------------------------
// Pass 1: diff = pred - target  (pure HBM streaming, float4)
// ---------------------------------------------------------------------------
__global__ void __launch_bounds__(256)
laplacian_diff_kernel(const float* __restrict__ pred,
                      const float* __restrict__ target,
                      float* __restrict__ diff, int n4) {
    int i = blockIdx.x * 256 + threadIdx.x;
    if (i < n4) {
        float4 a = reinterpret_cast<const float4*>(pred)[i];
        float4 b = reinterpret_cast<const float4*>(target)[i];
        float4 d;
        d.x = a.x - b.x; d.y = a.y - b.y; d.z = a.z - b.z; d.w = a.w - b.w;
        reinterpret_cast<float4*>(diff)[i] = d;
    }
}

// ---------------------------------------------------------------------------
// Pass 2: gather + Laplacian + |.| + block reduction.
// One wave32 per node n (all 4 batches); lane handles 2 features (32*2 = 64).
// Every neighbor-row load is a coalesced 256B transaction out of L2.
// USE_DIFF=true : src0 = precomputed diff.  false: src0=pred, src1=target.
// ---------------------------------------------------------------------------
template <bool USE_DIFF>
__global__ void __launch_bounds__(256)
laplacian_gather_kernel(const float* __restrict__ src0,
                        const float* __restrict__ src1,
                        const int* __restrict__ neighbors,
                        float* __restrict__ partials) {
    __shared__ float sdata[256];
    const int tid  = threadIdx.x;
    const int lane = tid & 31;
    const int wv   = tid >> 5;
    const int n    = blockIdx.x * 8 + wv;

    float sabs = 0.0f;
    if (n < NN) {
        // 16 neighbor indices, same 64B cacheline for all lanes of the wave.
        const int4* nb = reinterpret_cast<const int4*>(neighbors + (size_t)n * KK);
        int4 q0 = nb[0], q1 = nb[1], q2 = nb[2], q3 = nb[3];
        int idx[16] = {q0.x, q0.y, q0.z, q0.w, q1.x, q1.y, q1.z, q1.w,
                       q2.x, q2.y, q2.z, q2.w, q3.x, q3.y, q3.z, q3.w};

        #pragma unroll
        for (int b = 0; b < BB; ++b) {
            const size_t base = (size_t)b * NN * DD;
            const float* s0 = src0 + base;
            float2 own0 = *reinterpret_cast<const float2*>(s0 + (size_t)n * DD + 2 * lane);
            float2 acc0 = make_float2(0.0f, 0.0f);
            #pragma unroll
            for (int k = 0; k < KK; ++k) {
                float2 g = *reinterpret_cast<const float2*>(s0 + (size_t)idx[k] * DD + 2 * lane);
                acc0.x += g.x; acc0.y += g.y;
            }
            float vx, vy;
            if (USE_DIFF) {
                vx = own0.x - acc0.x * (1.0f / 16.0f);
                vy = own0.y - acc0.y * (1.0f / 16.0f);
            } else {
                const float* s1 = src1 + base;
                float2 own1 = *reinterpret_cast<const float2*>(s1 + (size_t)n * DD + 2 * lane);
                float2 acc1 = make_float2(0.0f, 0.0f);
                #pragma unroll
                for (int k = 0; k < KK; ++k) {
                    float2 g = *reinterpret_cast<const float2*>(s1 + (size_t)idx[k] * DD + 2 * lane);
                    acc1.x += g.x; acc1.y += g.y;
                }
                vx = (own0.x - own1.x) - (acc0.x - acc1.x) * (1.0f / 16.0f);
                vy = (own0.y - own1.y) - (acc0.y - acc1.y) * (1.0f / 16.0f);
            }
            sabs += fabsf(vx) + fabsf(vy);
        }
    }

    sdata[tid] = sabs;
    __syncthreads();
    #pragma unroll
    for (int s = 128; s > 0; s >>= 1) {
        if (tid < s) sdata[tid] += sdata[tid + s];
        __syncthreads();
    }
    if (tid == 0) partials[blockIdx.x] = sdata[0];
}

// ---------------------------------------------------------------------------
// Pass 3: final reduction of 6250 partials -> scalar mean.
// 256 strided accumulators -> LDS; wave 0 reduces all 256 values with four
// chained V_WMMA_F32_16X16X4_F32 (A = ones): D[m,n] = sum_k B[k,n] + C[m,n].
// Each B slot is a distinct LDS value, so the row-sum of D is the exact total
// (rows M=0..15 are replicated; lanes 16-31 duplicate lanes 0-15 -> *0.5).
// ---------------------------------------------------------------------------
__global__ void __launch_bounds__(256)
laplacian_final_kernel(const float* __restrict__ partials, int np,
                       float inv_count, float* __restrict__ out) {
    __shared__ float sdata[256];
    const int tid = threadIdx.x;
    float s = 0.0f;
    for (int i = tid; i < np; i += 256) s += partials[i];
    sdata[tid] = s;
    __syncthreads();

    if (tid < 32) {  // wave 0: EXEC is all ones here (uniform branch)
        float v;
#if __has_builtin(__builtin_amdgcn_wmma_f32_16x16x4_f32)
        v2f a = {1.0f, 1.0f};       // ones A-matrix (16x4)
        v8f c = {0.0f, 0.0f, 0.0f, 0.0f, 0.0f, 0.0f, 0.0f, 0.0f};
        #pragma unroll
        for (int i = 0; i < 4; ++i) {
            v2f bv;
            bv[0] = sdata[i * 64 + tid];
            bv[1] = sdata[i * 64 + 32 + tid];
            c = __builtin_amdgcn_wmma_f32_16x16x4_f32(
                    false, a, false, bv, (short)0, c, false, false);
        }
        v = c[0];                   // row M=0 (lanes 0-15), M=8 dup (16-31)
        #pragma unroll
        for (int off = 16; off > 0; off >>= 1) v += __shfl_xor(v, off);
        v *= 0.5f;                  // lanes 16-31 duplicated the row
#else
        v = 0.0f;
        if (tid == 0) for (int i = 0; i < 256; ++i) v += sdata[i];
#endif
        if (tid == 0) out[0] = v * inv_count;
    }
}

// ---------------------------------------------------------------------------
extern "C" void kernel_launch(void* const* d_in, const int* in_sizes, int n_in,
                              void* d_out, int out_size, void* d_ws, size_t ws_size,
                              hipStream_t stream) {
    const float* pred      = (const float*)d_in[0];
    const float* target    = (const float*)d_in[1];
    const int*   neighbors = (const int*)d_in[2];
    float*       out       = (float*)d_out;

    const size_t diffBytes = (size_t)BB * NN * DD * sizeof(float);  // 51.2 MB
    const int    nPart     = (NN + 7) / 8;                          // 6250 blocks
    const float  invCount  = 1.0f / (float)((size_t)BB * NN * DD);

    if (ws_size >= diffBytes + (size_t)nPart * sizeof(float)) {
        float* diff     = (float*)d_ws;
        float* partials = (float*)((char*)d_ws + diffBytes);
        const int n4 = BB * NN * DD / 4;
        laplacian_diff_kernel<<<(n4 + 255) / 256, 256, 0, stream>>>(pred, target, diff, n4);
        laplacian_gather_kernel<true><<<nPart, 256, 0, stream>>>(diff, nullptr, neighbors, partials);
        laplacian_final_kernel<<<1, 256, 0, stream>>>(partials, nPart, invCount, out);
    } else {
        // Fused fallback: gather pred & target directly (both live in L2).
        float* partials = (float*)d_ws;
        laplacian_gather_kernel<false><<<nPart, 256, 0, stream>>>(pred, target, neighbors, partials);
        laplacian_final_kernel<<<1, 256, 0, stream>>>(partials, nPart, invCount, out);
    }
}